// ScaledDotProductAttention_87634512708213
// MI455X (gfx1250) — compile-verified
//
#include <hip/hip_runtime.h>
#include <hip/hip_bf16.h>

#define NB 64
#define NT 4096
#define NC 10
#define NDV 256
#define CP 16          // C padded to 16 rows for WMMA M dimension

typedef __attribute__((ext_vector_type(2))) float v2f;
typedef __attribute__((ext_vector_type(8))) float v8f;

// -------------------------------------------------------------------------
// Kernel 1: per-batch masked softmax of logits s[c,t] = sum_j q[c,j]*k[b,t,j],
// mask applied multiplicatively to logits BEFORE softmax (matches reference).
// Writes P[b][c][t] row-major (c padded to 16, pad rows = 0) for WMMA A-loads.
// One block per batch, 256 threads, 3 streaming passes over k[b] (L2-resident).
// -------------------------------------------------------------------------
__global__ __launch_bounds__(256)
void attn_softmax_kernel(const float* __restrict__ k,
                         const float* __restrict__ q,
                         const int*   __restrict__ mask,
                         float* __restrict__ P) {
    __shared__ float qs[NC * NC];
    __shared__ float red[256];
    __shared__ float smax[NC];
    __shared__ float sinv[NC];

    const int b   = blockIdx.x;
    const int tid = threadIdx.x;

    if (tid < NC * NC) qs[tid] = q[tid];
    __syncthreads();

    const float* kb = k    + (size_t)b * NT * NC;
    const int*   mb = mask + (size_t)b * NT;

    // ---- Pass 1: per-c max of masked logits (masked -> logit 0 included) ----
    float lmax[NC];
#pragma unroll
    for (int c = 0; c < NC; ++c) lmax[c] = -3.402823466e38f;

    for (int t = tid; t < NT; t += 256) {
        float kr[NC];
#pragma unroll
        for (int j = 0; j < NC; ++j) kr[j] = kb[t * NC + j];
        const float m = (float)mb[t];
#pragma unroll
        for (int c = 0; c < NC; ++c) {
            float s = 0.f;
#pragma unroll
            for (int j = 0; j < NC; ++j) s = fmaf(qs[c * NC + j], kr[j], s);
            s *= m;
            lmax[c] = fmaxf(lmax[c], s);
        }
    }
#pragma unroll
    for (int c = 0; c < NC; ++c) {
        red[tid] = lmax[c];
        __syncthreads();
        for (int off = 128; off > 0; off >>= 1) {
            if (tid < off) red[tid] = fmaxf(red[tid], red[tid + off]);
            __syncthreads();
        }
        if (tid == 0) smax[c] = red[0];
        __syncthreads();
    }

    // ---- Pass 2: per-c sum of exp(s - max) ----
    float lsum[NC];
#pragma unroll
    for (int c = 0; c < NC; ++c) lsum[c] = 0.f;

    for (int t = tid; t < NT; t += 256) {
        float kr[NC];
#pragma unroll
        for (int j = 0; j < NC; ++j) kr[j] = kb[t * NC + j];
        const float m = (float)mb[t];
#pragma unroll
        for (int c = 0; c < NC; ++c) {
            float s = 0.f;
#pragma unroll
            for (int j = 0; j < NC; ++j) s = fmaf(qs[c * NC + j], kr[j], s);
            s *= m;
            lsum[c] += __expf(s - smax[c]);
        }
    }
#pragma unroll
    for (int c = 0; c < NC; ++c) {
        red[tid] = lsum[c];
        __syncthreads();
        for (int off = 128; off > 0; off >>= 1) {
            if (tid < off) red[tid] += red[tid + off];
            __syncthreads();
        }
        if (tid == 0) sinv[c] = 1.f / red[0];
        __syncthreads();
    }

    // ---- Pass 3: write normalized probabilities, zero pad rows ----
    float* Pb = P + (size_t)b * CP * NT;
    for (int t = tid; t < NT; t += 256) {
        float kr[NC];
#pragma unroll
        for (int j = 0; j < NC; ++j) kr[j] = kb[t * NC + j];
        const float m = (float)mb[t];
#pragma unroll
        for (int c = 0; c < NC; ++c) {
            float s = 0.f;
#pragma unroll
            for (int j = 0; j < NC; ++j) s = fmaf(qs[c * NC + j], kr[j], s);
            s *= m;
            Pb[(size_t)c * NT + t] = __expf(s - smax[c]) * sinv[c];
        }
#pragma unroll
        for (int c = NC; c < CP; ++c) Pb[(size_t)c * NT + t] = 0.f;
    }
}

// -------------------------------------------------------------------------
// Kernel 2: out[b, 0:10, n0:n0+16] = P[b] (16 x 4096) @ v[b] (4096 x 256) tile
// via chained V_WMMA_F32_16X16X4_F32. Grid: x = N-tile (16), y = batch (64).
// 8 waves per block each own a K-chunk of 512; partial 16x16 tiles reduced
// through LDS. v is streamed from HBM exactly once across the whole grid.
// -------------------------------------------------------------------------
__global__ __launch_bounds__(256)
void attn_pv_wmma_kernel(const float* __restrict__ P,
                         const float* __restrict__ v,
                         float* __restrict__ out) {
    const int ntile = blockIdx.x;        // 0..15  (DV/16 column tiles)
    const int b     = blockIdx.y;        // 0..63
    const int tid   = threadIdx.x;
    const int wave  = tid >> 5;          // 0..7
    const int lane  = tid & 31;
    const int lo16  = lane & 15;
    const int hi    = lane >> 4;         // 0 or 1
    const int n0    = ntile * 16;

    const float* Pb = P + (size_t)b * CP * NT;
    const float* vb = v + (size_t)b * NT * NDV;

    // A fragment base: lane holds A[m = lo16][kk + 2*hi] / [kk + 1 + 2*hi]
    const float* Arow = Pb + (size_t)lo16 * NT + 2 * hi;
    // B fragment base: lane holds B[kk + 2*hi][n0 + lo16] / [kk + 1 + 2*hi][...]
    const float* Bcol = vb + (size_t)(2 * hi) * NDV + n0 + lo16;

    v8f acc = {0.f, 0.f, 0.f, 0.f, 0.f, 0.f, 0.f, 0.f};

    const int kbeg = wave * (NT / 8);
    const int kend = kbeg + (NT / 8);

#pragma unroll 4
    for (int kk = kbeg; kk < kend; kk += 4) {
        v2f a = *(const v2f*)(Arow + kk);              // two consecutive K values
        v2f bm;
        bm.x = Bcol[(size_t)kk * NDV];
        bm.y = Bcol[(size_t)kk * NDV + NDV];
        // 8 args: (neg_a, A, neg_b, B, c_mod, C, reuse_a, reuse_b)
        acc = __builtin_amdgcn_wmma_f32_16x16x4_f32(
            false, a, false, bm, (short)0, acc, false, false);
    }

    // Reduce the 8 per-wave 16x16 partial tiles through LDS.
    __shared__ float tile[8][16][16];
#pragma unroll
    for (int i = 0; i < 8; ++i)
        tile[wave][i + 8 * hi][lo16] = acc[i];   // C/D layout: VGPR i -> M = i (+8 for hi lanes)
    __syncthreads();

    const int m = tid >> 4;      // 0..15
    const int n = tid & 15;      // 0..15
    float sum = 0.f;
#pragma unroll
    for (int w = 0; w < 8; ++w) sum += tile[w][m][n];

    if (m < NC)
        out[((size_t)b * NC + m) * NDV + n0 + n] = sum;
}

extern "C" void kernel_launch(void* const* d_in, const int* in_sizes, int n_in,
                              void* d_out, int out_size, void* d_ws, size_t ws_size,
                              hipStream_t stream) {
    const float* k    = (const float*)d_in[0];   // [B, T, C]
    const float* v    = (const float*)d_in[1];   // [B, T, DV]
    const float* q    = (const float*)d_in[2];   // [C, C]
    const int*   mask = (const int*)d_in[3];     // [B, T]
    float* out = (float*)d_out;                  // [B, C, DV]

    float* P = (float*)d_ws;                     // [B, 16, T] = 16 MB staging

    attn_softmax_kernel<<<dim3(NB), 256, 0, stream>>>(k, q, mask, P);
    attn_pv_wmma_kernel<<<dim3(NDV / 16, NB), 256, 0, stream>>>(P, v, out);
}